// Attention_21285857919204
// MI455X (gfx1250) — compile-verified
//
#include <hip/hip_runtime.h>
#include <math.h>

// Problem constants (B,H,S,D) = (4,8,512,64)
#define BB 4
#define HH 8
#define SS 512
#define DD 64

typedef __attribute__((ext_vector_type(2))) float v2f;
typedef __attribute__((ext_vector_type(8))) float v8f;

// D = A(16x4) * B(4x16) + C  -- f32 WMMA, wave32.
// A frag (v2f): lane<16 -> row M=lane, {K=k0,K=k0+1}; lane>=16 -> row M=lane-16, {K=k0+2,K=k0+3}
// B frag (v2f): lane<16 -> col N=lane, {K=k0,K=k0+1}; lane>=16 -> col N=lane-16, {K=k0+2,K=k0+3}
// C/D (v8f):   element r: lane<16 -> (M=r, N=lane); lane>=16 -> (M=r+8, N=lane-16)
static __device__ __forceinline__ v8f wmma_f32(v2f a, v2f b, v8f c) {
  return __builtin_amdgcn_wmma_f32_16x16x4_f32(false, a, false, b, (short)0, c,
                                               false, false);
}

// ---------------------------------------------------------------------------
// Kernel 1: scores[b,h,s,t] = sum_d q[b,h,s,d]*k[b,h,t,d]   (unscaled)
// One wave per 16x16 (s,t) tile. Grid = B*H*(S/16)*(S/16) = 32768.
// ---------------------------------------------------------------------------
__global__ __launch_bounds__(32) void qk_kernel(const float* __restrict__ q,
                                                const float* __restrict__ k,
                                                float* __restrict__ scores) {
  const int bid = blockIdx.x;
  const int tt = bid & 31;
  const int st = (bid >> 5) & 31;
  const int h  = (bid >> 10) & 7;
  const int b  = bid >> 13;

  const int lane = threadIdx.x;
  const int half = lane >> 4;   // 0: K 0..1, 1: K 2..3
  const int mrow = lane & 15;   // row (A) / col (B) within tile

  const float* qbase = q + (((size_t)(b * HH + h) * SS) + st * 16) * DD;
  const float* kbase = k + (((size_t)(b * HH + h) * SS) + tt * 16) * DD;

  v8f acc = {};
#pragma unroll
  for (int k0 = 0; k0 < DD; k0 += 4) {
    const int ka = k0 + half * 2;
    v2f a = *(const v2f*)(qbase + mrow * DD + ka);
    v2f bfrag = *(const v2f*)(kbase + mrow * DD + ka);
    acc = wmma_f32(a, bfrag, acc);
  }

  float* out = scores + ((size_t)((b * HH + h) * SS) + st * 16) * SS + tt * 16;
#pragma unroll
  for (int r = 0; r < 8; ++r) {
    out[(size_t)(r + half * 8) * SS + mrow] = acc[r];
  }
}

// ---------------------------------------------------------------------------
// Kernel 2: scores[b,h,s,t] += sum_d q[b,h,s,d]*relem[b,s,t,d]
// Per (b,s): GEMM with M = heads (8 real rows, 8 zero rows).
// Grid = B*S*(S/16) = 65536 waves.
// ---------------------------------------------------------------------------
__global__ __launch_bounds__(32) void qrel_kernel(const float* __restrict__ q,
                                                  const float* __restrict__ rel,
                                                  float* __restrict__ scores) {
  const int bid = blockIdx.x;
  const int tt = bid & 31;
  const int s  = (bid >> 5) & (SS - 1);
  const int b  = bid >> 14;

  const int lane = threadIdx.x;
  const int half = lane >> 4;
  const int mrow = lane & 15;

  // A rows = heads; rows 8..15 forced to zero (branchless, EXEC stays full).
  const float zmask = (mrow < HH) ? 1.0f : 0.0f;
  const float* qrow = q + (((size_t)(b * HH + (mrow & 7)) * SS) + s) * DD;
  const float* rbase = rel + (((size_t)b * SS + s) * SS + tt * 16) * DD;

  v8f acc = {};
#pragma unroll
  for (int k0 = 0; k0 < DD; k0 += 4) {
    const int ka = k0 + half * 2;
    v2f a = *(const v2f*)(qrow + ka);
    a = a * zmask;
    v2f bfrag = *(const v2f*)(rbase + mrow * DD + ka);  // B[k][n]=rel[t0+n][k]
    acc = wmma_f32(a, bfrag, acc);
  }

  // Rows 0..7 (the real heads) live in acc[0..7] of lanes 0..15.
  if (half == 0) {
#pragma unroll
    for (int r = 0; r < 8; ++r) {
      const size_t idx =
          (((size_t)(b * HH + r) * SS) + s) * SS + tt * 16 + mrow;
      scores[idx] += acc[r];
    }
  }
}

// ---------------------------------------------------------------------------
// Kernel 3: scale by 1/sqrt(D)=0.125, mask, softmax over t (in place).
// One wave per row of 512. Grid = B*H*S = 16384.
// ---------------------------------------------------------------------------
__global__ __launch_bounds__(32) void softmax_kernel(float* __restrict__ scores,
                                                     const int* __restrict__ mask) {
  const int row = blockIdx.x;                 // (b*H + h)*S + s
  const int b = row >> 12;                    // H*S = 4096
  const int lane = threadIdx.x;

  float* p = scores + (size_t)row * SS;
  const int* mrow = mask + b * SS;

  float vals[16];
  float mx = -INFINITY;
#pragma unroll
  for (int i = 0; i < 16; ++i) {
    const int t = lane + 32 * i;
    float v = p[t] * 0.125f;
    v = (mrow[t] == 0) ? -1.0e9f : v;
    vals[i] = v;
    mx = fmaxf(mx, v);
  }
#pragma unroll
  for (int off = 16; off > 0; off >>= 1)
    mx = fmaxf(mx, __shfl_xor(mx, off, 32));

  float sum = 0.0f;
#pragma unroll
  for (int i = 0; i < 16; ++i) {
    const float e = __expf(vals[i] - mx);
    vals[i] = e;
    sum += e;
  }
#pragma unroll
  for (int off = 16; off > 0; off >>= 1)
    sum += __shfl_xor(sum, off, 32);

  const float inv = 1.0f / sum;
#pragma unroll
  for (int i = 0; i < 16; ++i) p[lane + 32 * i] = vals[i] * inv;
}

// ---------------------------------------------------------------------------
// Kernel 4: ans1[b,h,s,d] = sum_t p[b,h,s,t]*v[b,h,t,d]
// One wave per (b,h,s-tile,d-tile); K = 512 -> 128 WMMAs. Grid = 4096.
// ---------------------------------------------------------------------------
__global__ __launch_bounds__(32) void pv_kernel(const float* __restrict__ scores,
                                                const float* __restrict__ val,
                                                float* __restrict__ out1) {
  const int bid = blockIdx.x;
  const int dt = bid & 3;
  const int st = (bid >> 2) & 31;
  const int h  = (bid >> 7) & 7;
  const int b  = bid >> 10;

  const int lane = threadIdx.x;
  const int half = lane >> 4;
  const int mrow = lane & 15;

  const float* pbase = scores + ((size_t)((b * HH + h) * SS) + st * 16) * SS;
  const float* vbase = val + ((size_t)(b * HH + h) * SS) * DD + dt * 16;

  v8f acc = {};
#pragma unroll 8
  for (int k0 = 0; k0 < SS; k0 += 4) {
    const int ka = k0 + half * 2;
    v2f a = *(const v2f*)(pbase + (size_t)mrow * SS + ka);  // A[m][k] = P[s0+m][k]
    v2f bfrag = {vbase[(size_t)(ka + 0) * DD + mrow],       // B[k][n] = V[t=k][d0+n]
                 vbase[(size_t)(ka + 1) * DD + mrow]};
    acc = wmma_f32(a, bfrag, acc);
  }

  float* out = out1 + ((size_t)((b * HH + h) * SS) + st * 16) * DD + dt * 16;
#pragma unroll
  for (int r = 0; r < 8; ++r) {
    out[(size_t)(r + half * 8) * DD + mrow] = acc[r];
  }
}

// ---------------------------------------------------------------------------
// Kernel 5: ans[b,h,s,d] += sum_t p[b,h,s,t]*relem[b,s,t,d]
// Per (b,s,d-tile): GEMM with M = heads (8 real rows). Grid = 8192.
// ---------------------------------------------------------------------------
__global__ __launch_bounds__(32) void prel_kernel(const float* __restrict__ scores,
                                                  const float* __restrict__ rel,
                                                  float* __restrict__ out1) {
  const int bid = blockIdx.x;
  const int dt = bid & 3;
  const int s  = (bid >> 2) & (SS - 1);
  const int b  = bid >> 11;

  const int lane = threadIdx.x;
  const int half = lane >> 4;
  const int mrow = lane & 15;

  const float zmask = (mrow < HH) ? 1.0f : 0.0f;
  const float* prow = scores + (((size_t)(b * HH + (mrow & 7)) * SS) + s) * SS;
  const float* rbase = rel + (((size_t)b * SS + s) * SS) * DD + dt * 16;

  v8f acc = {};
#pragma unroll 8
  for (int k0 = 0; k0 < SS; k0 += 4) {
    const int ka = k0 + half * 2;
    v2f a = *(const v2f*)(prow + ka);                   // A[m][k] = P[h=m][k]
    a = a * zmask;
    v2f bfrag = {rbase[(size_t)(ka + 0) * DD + mrow],  // B[k][n] = rel[t=k][d0+n]
                 rbase[(size_t)(ka + 1) * DD + mrow]};
    acc = wmma_f32(a, bfrag, acc);
  }

  if (half == 0) {
#pragma unroll
    for (int r = 0; r < 8; ++r) {
      const size_t idx =
          (((size_t)(b * HH + r) * SS) + s) * DD + dt * 16 + mrow;
      out1[idx] += acc[r];
    }
  }
}

// ---------------------------------------------------------------------------
extern "C" void kernel_launch(void* const* d_in, const int* in_sizes, int n_in,
                              void* d_out, int out_size, void* d_ws, size_t ws_size,
                              hipStream_t stream) {
  const float* q    = (const float*)d_in[0];
  const float* k    = (const float*)d_in[1];
  const float* v    = (const float*)d_in[2];
  const float* rel  = (const float*)d_in[3];
  const int*   mask = (const int*)d_in[4];

  float* out1   = (float*)d_out;                        // (B,H,S,D) ans1+ans2
  float* scores = out1 + (size_t)BB * HH * SS * DD;     // (B,H,S,S) p_attn

  // Ordered on one stream: raw QK^T -> +Q.rel^T -> softmax -> P.V -> +P.rel
  qk_kernel<<<BB * HH * (SS / 16) * (SS / 16), 32, 0, stream>>>(q, k, scores);
  qrel_kernel<<<BB * SS * (SS / 16), 32, 0, stream>>>(q, rel, scores);
  softmax_kernel<<<BB * HH * SS, 32, 0, stream>>>(scores, mask);
  pv_kernel<<<BB * HH * (SS / 16) * 4, 32, 0, stream>>>(scores, v, out1);
  prel_kernel<<<BB * SS * 4, 32, 0, stream>>>(scores, rel, out1);
}